// Multinomial_13752485281938
// MI455X (gfx1250) — compile-verified
//
#include <hip/hip_runtime.h>
#include <hip/hip_bf16.h>
#include <stdint.h>

#define CAP   (1 << 21)   // number of leaves (power of two -> complete tree)
#define DEPTH 21          // log2(CAP) traversal steps
#define LDS_NODES 8192    // stage tree[0 .. 8192) : levels 0..12 (+1 node)

typedef __attribute__((address_space(3))) float lds_f32;

// ---------------------------------------------------------------------------
// K1: build leaves + lower interior levels.
// Each 256-thread block owns 2048 leaves; reduces 11 levels in LDS, writing
// every produced level to its slot in the global tree. Produces the level of
// 1024 nodes (offset 1023) as its final output (one node per block).
// ---------------------------------------------------------------------------
__global__ void __launch_bounds__(256) build_lower(const float* __restrict__ prio,
                                                   float* __restrict__ tree) {
    __shared__ float bufA[2048];
    __shared__ float bufB[1024];
    const int t    = threadIdx.x;
    const int base = blockIdx.x * 2048;

    // Load leaves (coalesced), mirror into tree leaf slots.
#pragma unroll
    for (int k = 0; k < 8; ++k) {
        float v = prio[base + k * 256 + t];
        bufA[k * 256 + t] = v;
        tree[(CAP - 1) + base + k * 256 + t] = v;
    }
    __syncthreads();

    float* src = bufA;
    float* dst = bufB;
    int levelSize = CAP / 2;   // global node count of level being produced
    int c = 1024;              // this block's node count in that level
    while (c >= 1) {
        const int lbase = blockIdx.x * c;
        for (int j = t; j < c; j += 256) {
            float v = src[2 * j] + src[2 * j + 1];   // same pairing as reference
            dst[j] = v;
            tree[(levelSize - 1) + lbase + j] = v;
        }
        __syncthreads();
        float* tmp = src; src = dst; dst = tmp;
        levelSize >>= 1;
        c >>= 1;
    }
}

// ---------------------------------------------------------------------------
// K2: finish top of tree. Single block reduces the 1024-node level (offset
// 1023) through 10 more levels down to the root at tree[0].
// ---------------------------------------------------------------------------
__global__ void __launch_bounds__(512) build_upper(float* __restrict__ tree) {
    __shared__ float bufA[1024];
    __shared__ float bufB[512];
    const int t = threadIdx.x;
    for (int j = t; j < 1024; j += 512) bufA[j] = tree[1023 + j];
    __syncthreads();

    float* src = bufA;
    float* dst = bufB;
    int levelSize = 512;
    int c = 512;
    while (c >= 1) {
        for (int j = t; j < c; j += 512) {
            float v = src[2 * j] + src[2 * j + 1];
            dst[j] = v;
            tree[(levelSize - 1) + j] = v;
        }
        __syncthreads();
        float* tmp = src; src = dst; dst = tmp;
        levelSize >>= 1;
        c >>= 1;
    }
}

// ---------------------------------------------------------------------------
// K3: stratified traversal. Block stages the top 13 tree levels (32 KB) into
// LDS with CDNA5 async global->LDS loads, then each thread walks 4 samples
// (4 independent dependency chains for latency hiding). First 12 steps read
// LDS; remaining 9 read global (entire 16 MB tree is L2-resident; stratified
// samples make the gathers near-coalesced across a wave).
// ---------------------------------------------------------------------------
__global__ void __launch_bounds__(256) sample_kernel(const float* __restrict__ uniform,
                                                     const float* __restrict__ tree,
                                                     int* __restrict__ out,
                                                     int n) {
    __shared__ float s_tree[LDS_NODES];
    const int t = threadIdx.x;

    // Raw 32-bit LDS byte offset of s_tree for the async-load VDST operand.
    unsigned ldsBase = (unsigned)(size_t)(lds_f32*)&s_tree[0];

    // Stage tree[0 .. 8192): 8 rounds of 16B/lane async copies (ASYNCcnt).
#pragma unroll
    for (int it = 0; it < 8; ++it) {
        unsigned     loff = ldsBase + (unsigned)((it * 1024 + t * 4) * 4);
        const float* gp   = tree + (it * 1024 + t * 4);
        asm volatile("global_load_async_to_lds_b128 %0, %1, off"
                     :: "v"(loff), "v"(gp) : "memory");
    }
    asm volatile("s_wait_asynccnt 0x0" ::: "memory");
    __syncthreads();

    const float total = s_tree[0];
    const float step  = total / (float)n;          // linspace step
    const int   sampBase = blockIdx.x * 1024 + t;  // 4 samples per thread

    int   idx[4];
    float s[4];
#pragma unroll
    for (int k = 0; k < 4; ++k) {
        const int i = sampBase + k * 256;
        s[k]   = (uniform[i] * total) / (float)n + (float)i * step;
        idx[k] = 0;
    }

    // Steps 0..11: node index 'left' < 8191 -> guaranteed in LDS.
#pragma unroll
    for (int d = 0; d < 12; ++d) {
#pragma unroll
        for (int k = 0; k < 4; ++k) {
            const int   left = 2 * idx[k] + 1;
            const float lv   = s_tree[left];
            const bool  gl   = (s[k] <= lv);
            idx[k] = gl ? left : (left + 1);
            s[k]   = gl ? s[k] : (s[k] - lv);
        }
    }
    // Steps 12..20: global gathers (L2-resident tree, near-coalesced).
#pragma unroll
    for (int d = 12; d < DEPTH; ++d) {
#pragma unroll
        for (int k = 0; k < 4; ++k) {
            const int   left = 2 * idx[k] + 1;
            const float lv   = tree[left];
            const bool  gl   = (s[k] <= lv);
            idx[k] = gl ? left : (left + 1);
            s[k]   = gl ? s[k] : (s[k] - lv);
        }
    }

#pragma unroll
    for (int k = 0; k < 4; ++k) {
        const int i = sampBase + k * 256;
        out[i] = idx[k] - (CAP - 1);   // leaf index in [0, CAP)
    }
}

// ---------------------------------------------------------------------------
extern "C" void kernel_launch(void* const* d_in, const int* in_sizes, int n_in,
                              void* d_out, int out_size, void* d_ws, size_t ws_size,
                              hipStream_t stream) {
    const float* prio = (const float*)d_in[0];   // priorities [CAP] fp32
    const float* uni  = (const float*)d_in[1];   // uniform [SIZE] fp32
    const int    n    = in_sizes[1];             // SIZE = 1<<20

    float* tree = (float*)d_ws;                  // 2*CAP-1 floats (~16 MB)

    build_lower<<<CAP / 2048, 256, 0, stream>>>(prio, tree);
    build_upper<<<1, 512, 0, stream>>>(tree);
    sample_kernel<<<n / 1024, 256, 0, stream>>>(uni, tree, (int*)d_out, n);
}